// DeepPatchEncoder_58884001628673
// MI455X (gfx1250) — compile-verified
//
#include <hip/hip_runtime.h>
#include <hip/hip_bf16.h>

// ---------------------------------------------------------------------------
// Problem constants (from the reference)
// ---------------------------------------------------------------------------
#define BB   128
#define IMG  224
#define CC   3
#define P0   16
#define P1   8
#define N0   196           // (224/16)^2
#define N1   784           // (224/8)^2
#define D0   768           // 3*16*16
#define GM   192           // GEMM M = C * 8 * 8   (c,y,x)
#define GK   784           // GEMM K = 2*2*196     (ky,kx,i)
#define GN   784           // GEMM N = N1          (o)
#define BN_EPS 1e-3f

typedef __attribute__((ext_vector_type(2))) float v2f;
typedef __attribute__((ext_vector_type(8))) float v8f;

// ---------------------------------------------------------------------------
// Kernel 1: gather A into K-major layout A_T[k][m], m=(c*64 + y*8 + x),
// k=(ky*2+kx)*196 + i.   A[m,k] = pos_emb[i, ((2y+ky)*16 + (2x+kx))*3 + c]
// ---------------------------------------------------------------------------
__global__ void pack_a_kernel(const float* __restrict__ pos_emb,
                              float* __restrict__ A_T) {
    int t = blockIdx.x * blockDim.x + threadIdx.x;
    if (t >= GM * GK) return;
    int k = t / GM;            // K-major for coalesced GEMM loads
    int m = t - k * GM;
    int c = m >> 6;            // m / 64
    int y = (m >> 3) & 7;
    int x = m & 7;
    int g  = k / 196;          // (ky*2 + kx)
    int i  = k - g * 196;
    int ky = g >> 1;
    int kx = g & 1;
    int col = ((2 * y + ky) * 16 + (2 * x + kx)) * 3 + c;
    A_T[t] = pos_emb[i * D0 + col];
}

// ---------------------------------------------------------------------------
// Kernel 2: GEMM C[192,784] = A(192x784) * B(784x784) with V_WMMA_F32_16X16X4,
// fused inference-BatchNorm epilogue. One wave (32 lanes) per 16x16 tile.
// grid = (12, 49), block = 32.
//
// Fragment layouts (CDNA5 ISA 7.12.2, wave32):
//   A frag a[j], lane L : A[m = L&15, k = k0 + 2*(L>>4) + j]
//   B frag b[j], lane L : B[k = k0 + 2*(L>>4) + j, n = L&15]
//   C frag c[j], lane L : C[m = j + 8*(L>>4), n = L&15]
// ---------------------------------------------------------------------------
__global__ void conv_gemm_bn_kernel(const float* __restrict__ A_T,     // [GK][GM]
                                    const float* __restrict__ conv_w,  // [GK][GN]
                                    const float* __restrict__ bn_gamma,
                                    const float* __restrict__ bn_beta,
                                    const float* __restrict__ bn_mean,
                                    const float* __restrict__ bn_var,
                                    float* __restrict__ G_bn) {        // [GM][GN]
    const int lane  = threadIdx.x;        // 0..31
    const int mtile = blockIdx.x;         // 0..11
    const int ntile = blockIdx.y;         // 0..48
    const int row   = lane & 15;
    const int khalf = lane >> 4;          // 0 or 1

    const int mcol = mtile * 16 + row;    // A column (M index) for this lane
    const int ncol = ntile * 16 + row;    // B/N column for this lane

    v8f acc = {};
    #pragma unroll 4
    for (int k0 = 0; k0 < GK; k0 += 4) {
        int ka = k0 + 2 * khalf;
        v2f a, b;
        a.x = A_T[(size_t)(ka + 0) * GM + mcol];
        a.y = A_T[(size_t)(ka + 1) * GM + mcol];
        b.x = conv_w[(size_t)(ka + 0) * GN + ncol];
        b.y = conv_w[(size_t)(ka + 1) * GN + ncol];
        acc = __builtin_amdgcn_wmma_f32_16x16x4_f32(
            /*neg_a=*/false, a, /*neg_b=*/false, b,
            /*c_mod=*/(short)0, acc, /*reuse_a=*/false, /*reuse_b=*/false);
    }

    // Fused BN (inference): (x - mean) * gamma/sqrt(var+eps) + beta, per output channel o=n
    const int o = ncol;
    float scale = bn_gamma[o] * rsqrtf(bn_var[o] + BN_EPS);
    float shift = bn_beta[o] - bn_mean[o] * scale;

    #pragma unroll
    for (int j = 0; j < 8; ++j) {
        int m = mtile * 16 + j + 8 * khalf;
        G_bn[(size_t)m * GN + o] = acc[j] * scale + shift;
    }
}

// ---------------------------------------------------------------------------
// Kernel 3: assemble batch-invariant bias[196,768]:
//   bias[n0,d0] = pos_emb[n0,d0] + G_bn[(c*64 + (h%8)*8 + (w%8)) * 784 + (h/8)*28 + (w/8)]
// with n0=(py,px), d0=(r,s,c), h=py*16+r, w=px*16+s.
// ---------------------------------------------------------------------------
__global__ void bias_kernel(const float* __restrict__ pos_emb,
                            const float* __restrict__ G_bn,
                            float* __restrict__ bias) {
    int t = blockIdx.x * blockDim.x + threadIdx.x;
    if (t >= N0 * D0) return;
    int n0 = t / D0;
    int d0 = t - n0 * D0;
    int py = n0 / 14, px = n0 - py * 14;
    int r  = d0 / 48;
    int rem = d0 - r * 48;
    int s  = rem / 3;
    int c  = rem - s * 3;
    int h  = py * 16 + r;
    int w  = px * 16 + s;
    int m  = c * 64 + (h & 7) * 8 + (w & 7);
    int o  = (h >> 3) * 28 + (w >> 3);
    bias[t] = pos_emb[t] + G_bn[(size_t)m * GN + o];
}

// ---------------------------------------------------------------------------
// Kernel 4 (dominant, pure streaming): out[b, j] = X[b, perm(j)] + bias[j]
// j = n0*768 + d0 ; X index = ((b*224 + h)*224)*3 + px*48 + rem
// 4 consecutive floats per thread: float4 store (16B aligned), float4 bias
// load (L2-resident), 4 scalar X loads (48-float contiguous segments).
// ---------------------------------------------------------------------------
__global__ void stream_add_kernel(const float* __restrict__ X,
                                  const float* __restrict__ bias,
                                  float* __restrict__ out) {
    const int J  = N0 * D0;      // 150528 floats per batch image
    const int J4 = J / 4;        // 37632 quads per batch image
    long long tid = (long long)blockIdx.x * blockDim.x + threadIdx.x;
    if (tid >= (long long)BB * J4) return;

    int b = (int)(tid / J4);
    int q = (int)(tid - (long long)b * J4);
    int j = q * 4;

    int n0 = j / D0;
    int d0 = j - n0 * D0;
    int py = n0 / 14, px = n0 - py * 14;
    int r   = d0 / 48;
    int rem = d0 - r * 48;            // multiple of 4, stays within one 48-seg
    int h   = py * 16 + r;

    size_t xbase = ((size_t)b * IMG + h) * (IMG * CC) + (size_t)px * 48 + rem;

    float4 bv = *(const float4*)(bias + j);
    float4 ov;
    ov.x = X[xbase + 0] + bv.x;
    ov.y = X[xbase + 1] + bv.y;
    ov.z = X[xbase + 2] + bv.z;
    ov.w = X[xbase + 3] + bv.w;
    *(float4*)(out + (size_t)b * J + j) = ov;
}

// ---------------------------------------------------------------------------
// Launch
// ---------------------------------------------------------------------------
extern "C" void kernel_launch(void* const* d_in, const int* in_sizes, int n_in,
                              void* d_out, int out_size, void* d_ws, size_t ws_size,
                              hipStream_t stream) {
    const float* X        = (const float*)d_in[0];
    const float* pos_emb  = (const float*)d_in[1];
    const float* conv_w   = (const float*)d_in[2];
    const float* bn_gamma = (const float*)d_in[3];
    const float* bn_beta  = (const float*)d_in[4];
    const float* bn_mean  = (const float*)d_in[5];
    const float* bn_var   = (const float*)d_in[6];
    float* out = (float*)d_out;

    // Workspace layout (all 16B-aligned: each region is 602112 bytes)
    float* A_T  = (float*)d_ws;                       // [784][192]
    float* G_bn = A_T  + (size_t)GM * GK;             // [192][784]
    float* bias = G_bn + (size_t)GM * GN;             // [196][768]

    // 1) pack A (K-major)
    {
        int n = GM * GK;
        pack_a_kernel<<<(n + 255) / 256, 256, 0, stream>>>(pos_emb, A_T);
    }
    // 2) WMMA f32 GEMM + fused BN: one wave per 16x16 tile
    {
        dim3 grid(GM / 16, GN / 16);   // (12, 49)
        conv_gemm_bn_kernel<<<grid, 32, 0, stream>>>(A_T, conv_w, bn_gamma,
                                                     bn_beta, bn_mean, bn_var,
                                                     G_bn);
    }
    // 3) batch-invariant bias
    {
        int n = N0 * D0;
        bias_kernel<<<(n + 255) / 256, 256, 0, stream>>>(pos_emb, G_bn, bias);
    }
    // 4) dominant streaming pass: 128 * 37632 quad-threads
    {
        long long n = (long long)BB * (N0 * D0 / 4);
        int blocks = (int)((n + 255) / 256);
        stream_add_kernel<<<blocks, 256, 0, stream>>>(X, bias, out);
    }
}